// sfr_13907104105211
// MI455X (gfx1250) — compile-verified
//
#include <hip/hip_runtime.h>
#include <hip/hip_bf16.h>
#include <stdint.h>

#define BATCH_N   16384
#define T_N       29            // round(0.14/0.005)+1
#define NSEG      3
#define G_TOTAL   (NSEG * T_N)  // 87 output rows
#define L0_C      0.1f
#define D_C       0.0075f
#define DS_C      0.005f

struct RodState {
    float r0, r1, r2;
    float R[9];     // row-major 3x3
};

// Advance state by arc length s along a constant-curvature arc with
// curvature omega = (ux, uy, 0):  R <- R*M(s),  r <- r + R*p(s).
__device__ __forceinline__ void advance_state(RodState& st, float ux, float uy, float s)
{
    float th2 = ux * ux + uy * uy;
    float th  = sqrtf(th2);
    float ang = s * th;
    float sn  = sinf(ang);
    float cn  = cosf(ang);
    float f, g;
    if (th2 > 1e-12f) {
        f = sn / th;                 // sin(s*th)/th
        g = (1.0f - cn) / th2;       // (1-cos(s*th))/th^2
    } else {
        f = s;                       // limits as th -> 0
        g = 0.5f * s * s;
    }

    // M = I + f*K + g*K^2,  K = [[0,0,uy],[0,0,-ux],[-uy,ux,0]]
    const float M00 = 1.0f - uy * uy * g;
    const float M01 = ux * uy * g;
    const float M02 = uy * f;
    const float M10 = M01;
    const float M11 = 1.0f - ux * ux * g;
    const float M12 = -ux * f;
    const float M20 = -uy * f;
    const float M21 = ux * f;
    const float M22 = cn;

    // p = integral of M(t)*e3 dt = (uy*g, -ux*g, f)
    const float p0 = uy * g;
    const float p1 = -ux * g;
    const float p2 = f;

    // r += R * p
    const float nr0 = st.r0 + st.R[0] * p0 + st.R[1] * p1 + st.R[2] * p2;
    const float nr1 = st.r1 + st.R[3] * p0 + st.R[4] * p1 + st.R[5] * p2;
    const float nr2 = st.r2 + st.R[6] * p0 + st.R[7] * p1 + st.R[8] * p2;

    // R = R * M
    float N[9];
    N[0] = st.R[0] * M00 + st.R[1] * M10 + st.R[2] * M20;
    N[1] = st.R[0] * M01 + st.R[1] * M11 + st.R[2] * M21;
    N[2] = st.R[0] * M02 + st.R[1] * M12 + st.R[2] * M22;
    N[3] = st.R[3] * M00 + st.R[4] * M10 + st.R[5] * M20;
    N[4] = st.R[3] * M01 + st.R[4] * M11 + st.R[5] * M21;
    N[5] = st.R[3] * M02 + st.R[4] * M12 + st.R[5] * M22;
    N[6] = st.R[6] * M00 + st.R[7] * M10 + st.R[8] * M20;
    N[7] = st.R[6] * M01 + st.R[7] * M11 + st.R[8] * M21;
    N[8] = st.R[6] * M02 + st.R[7] * M12 + st.R[8] * M22;

    st.r0 = nr0; st.r1 = nr1; st.r2 = nr2;
#pragma unroll
    for (int i = 0; i < 9; ++i) st.R[i] = N[i];
}

__device__ __forceinline__ void seg_params(const float* a, int n,
                                           float& ux, float& uy, int& len)
{
    const float ext = a[3 * n + 0];
    const float by  = a[3 * n + 1];
    const float bx  = a[3 * n + 2];
    const float l   = L0_C + ext;
    const float inv = 1.0f / (l * D_C);
    ux  = -bx * inv;                 // a2 / -(l*D)
    uy  =  by * inv;                 // a1 /  (l*D)
    len = (int)(l / DS_C);           // float32 div + trunc, as in reference
}

// One thread per (global output row g, batch b). 64 blocks per g, so g is
// uniform per block. Each wave stages its 32 lanes' 14-float records into LDS
// (contiguous, output order) then fires async LDS->global stores.
__global__ void __launch_bounds__(256)
rod_rows_kernel(const float* __restrict__ actions, float* __restrict__ out)
{
    const int g    = blockIdx.x >> 6;          // 0 .. 86
    const int bblk = blockIdx.x & 63;          // 64 blocks of 256 per row
    const int b    = bblk * 256 + threadIdx.x; // 0 .. 16383
    const int seg  = g / T_N;
    const int row  = g - seg * T_N;

    float a[9];
#pragma unroll
    for (int i = 0; i < 9; ++i) a[i] = actions[b * 9 + i];

    RodState st;
    st.r0 = 0.0f; st.r1 = 0.0f; st.r2 = 0.0f;
#pragma unroll
    for (int i = 0; i < 9; ++i) st.R[i] = 0.0f;
    st.R[0] = 1.0f; st.R[4] = 1.0f; st.R[8] = 1.0f;

    float ux = 0.0f, uy = 0.0f;
    // Compose end transforms of preceding segments, then this segment's row.
    for (int m = 0; m <= seg; ++m) {
        int len;
        seg_params(a, m, ux, uy, len);
        const int   ti = (m < seg) ? len : ((row < len) ? row : len);
        const float s  = (float)ti * DS_C;
        advance_state(st, ux, uy, s);
    }

    float vals[14];
    vals[0] = st.r0; vals[1] = st.r1; vals[2] = st.r2;
#pragma unroll
    for (int i = 0; i < 9; ++i) vals[3 + i] = st.R[i];
    vals[12] = ux; vals[13] = uy;

    // ---- stage into LDS in exact output order (per-wave 1792 B tile) ----
    __shared__ __align__(64) float tile[8 * 448];
    const int lane = threadIdx.x & 31;
    const int wave = threadIdx.x >> 5;
    float* wbuf = tile + wave * 448;
    float* dst  = wbuf + lane * 14;            // 56 B stride: 8 B aligned
#pragma unroll
    for (int c = 0; c < 14; ++c) dst[c] = vals[c];

    // LDS writes must be visible to the async-store engine.
    asm volatile("s_wait_dscnt 0x0" ::: "memory");

    // ---- async LDS -> global: 3 x b128 (512 B/wave) + 1 x b64 (256 B) ----
    const uint32_t lwave = (uint32_t)(uintptr_t)wbuf;  // low 32 bits = LDS offset
    float* gwave = out + ((size_t)g * BATCH_N + (size_t)bblk * 256 + (size_t)wave * 32) * 14;
    const uint64_t gw = (uint64_t)(uintptr_t)gwave;    // 256 B aligned chunk base

#pragma unroll
    for (int k = 0; k < 3; ++k) {
        uint64_t ga = gw + (uint64_t)(k * 512 + lane * 16);
        uint32_t la = lwave + (uint32_t)(k * 512 + lane * 16);
        asm volatile("global_store_async_from_lds_b128 %0, %1, off"
                     :: "v"(ga), "v"(la) : "memory");
    }
    {
        uint64_t ga = gw + (uint64_t)(1536 + lane * 8);
        uint32_t la = lwave + (uint32_t)(1536 + lane * 8);
        asm volatile("global_store_async_from_lds_b64 %0, %1, off"
                     :: "v"(ga), "v"(la) : "memory");
    }
    asm volatile("s_wait_asynccnt 0x0" ::: "memory");
}

extern "C" void kernel_launch(void* const* d_in, const int* in_sizes, int n_in,
                              void* d_out, int out_size, void* d_ws, size_t ws_size,
                              hipStream_t stream)
{
    const float* actions = (const float*)d_in[0];
    float* out = (float*)d_out;

    const int blocks = G_TOTAL * (BATCH_N / 256);  // 87 * 64 = 5568
    rod_rows_kernel<<<dim3(blocks), dim3(256), 0, stream>>>(actions, out);
}